// AssociativeLIF_18614388261695
// MI455X (gfx1250) — compile-verified
//
#include <hip/hip_runtime.h>
#include <stdint.h>

// LIF scan: i_syn[t] = 0.5*i_syn[t-1] + x[t]
//           v[t]     = tau*v[t-1] + (1-tau)*i_syn[t]
//           spike[t] = v[t] >= thr
// Memory-bound (0.65 flop/byte): stream x via TDM (tensor_load_to_lds) into a
// 4-deep LDS ring, scan in registers, nontemporal streaming stores.

#define TCHUNK 16   // time steps staged per TDM tile
#define COLS   256  // neurons per block (== blockDim.x)
#define NBUF   4    // LDS ring depth (4 x 16KB = 64KB of 320KB WGP LDS)

typedef unsigned int u32x4 __attribute__((ext_vector_type(4)));
typedef int          i32x4 __attribute__((ext_vector_type(4)));
typedef int          i32x8 __attribute__((ext_vector_type(8)));

// Issue one TDM 2D tile load: COLS x TCHUNK f32 elements, row stride
// ROW_STRIDE elements, from global gsrc into LDS byte offset lds_addr.
// Descriptor bit layout per CDNA5 ISA ch.8 (D# group0 / group1).
template <int ROW_STRIDE>
__device__ __forceinline__ void tdm_load_tile(unsigned lds_addr,
                                              const float* gsrc) {
  unsigned long long ga = (unsigned long long)(uintptr_t)gsrc;
  u32x4 g0;
  g0[0] = 1u;                                               // count=1, user desc
  g0[1] = lds_addr;                                         // lds_addr (bytes)
  g0[2] = (unsigned)ga;                                     // global_addr[31:0]
  g0[3] = ((unsigned)(ga >> 32) & 0x01FFFFFFu)              // global_addr[56:32]
          | 0x80000000u;                                    // type=2 ("image")
  i32x8 g1;                             // fully compile-time constant now
  g1[0] = 0x00020000;                   // workgroup_mask=0, data_size=2 (4B)
  g1[1] = (int)(COLS << 16);            // tensor_dim0 lo16 in bits[63:48]
  g1[2] = (int)(TCHUNK << 16);          // tensor_dim0 hi=0, tensor_dim1 lo
  g1[3] = (int)(COLS << 16);            // tensor_dim1 hi=0, tile_dim0=COLS
  g1[4] = TCHUNK;                       // tile_dim1=TCHUNK, tile_dim2=0
  g1[5] = ROW_STRIDE;                   // tensor_dim0_stride[31:0] (elements)
  g1[6] = 0;                            // stride hi / dim1_stride lo
  g1[7] = 0;
  i32x4 z4 = {0, 0, 0, 0};
#if __clang_major__ >= 23
  i32x8 z8 = {0, 0, 0, 0, 0, 0, 0, 0};
  __builtin_amdgcn_tensor_load_to_lds(g0, g1, z4, z4, z8, 0);
#else
  __builtin_amdgcn_tensor_load_to_lds(g0, g1, z4, z4, 0);
#endif
}

template <int T, int N>
__global__ __launch_bounds__(COLS) void AssociativeLIF_kernel(
    const float* __restrict__ x,
    const float* __restrict__ tau_mem,
    const float* __restrict__ v_threshold,
    float* __restrict__ out) {
  __shared__ float sbuf[NBUF][TCHUNK * COLS];  // 4 x 16KB ring

  constexpr int C = T / TCHUNK;                // chunks per block (8)
  static_assert(C >= NBUF, "pipeline assumes C >= NBUF");

  const int tid = threadIdx.x;
  const int b   = blockIdx.y;            // batch
  const int n0  = blockIdx.x * COLS;     // neuron tile base
  const int n   = n0 + tid;

  float tau = tau_mem[n];
  tau = fminf(fmaxf(tau, 0.8f), 0.98f);
  float thr = v_threshold[n];
  thr = fminf(fmaxf(thr, 0.05f), 0.5f);
  const float omt = 1.0f - tau;

  const size_t base = (size_t)b * (size_t)(T * N) + (size_t)n0;
  const bool lead   = (tid < 32);        // wave 0 drives the TDM pipeline

  // Prologue: fill the ring (NBUF tiles in flight).
  if (lead) {
#pragma unroll
    for (int p = 0; p < NBUF; ++p) {
      tdm_load_tile<N>((unsigned)(uintptr_t)&sbuf[p][0],
                       x + base + (size_t)(p * TCHUNK * N));
    }
  }

  float isyn = 0.0f, v = 0.0f;
  for (int c = 0; c < C; ++c) {
    if (lead) {
      // Chunks complete in order; allow (#chunks issued after c) outstanding.
      if (c + NBUF - 1 < C)      __builtin_amdgcn_s_wait_tensorcnt(NBUF - 1);
      else if (c + NBUF - 2 < C) __builtin_amdgcn_s_wait_tensorcnt(NBUF - 2);
      else if (c + NBUF - 3 < C) __builtin_amdgcn_s_wait_tensorcnt(NBUF - 3);
      else                       __builtin_amdgcn_s_wait_tensorcnt(0);
    }
    __syncthreads();  // publish staged tile to all 8 waves

    const float* tile = sbuf[c & (NBUF - 1)];
    size_t orow = base + (size_t)c * (TCHUNK * N) + (size_t)tid;
#pragma unroll
    for (int r = 0; r < TCHUNK; ++r) {
      float xv = tile[r * COLS + tid];
      isyn = fmaf(0.5f, isyn, xv);          // synaptic current scan
      v    = fmaf(tau, v, omt * isyn);      // membrane scan
      float sp = (v >= thr) ? 1.0f : 0.0f;  // Heaviside spike
      __builtin_nontemporal_store(sp, &out[orow]);
      orow += (size_t)N;
    }

    __syncthreads();  // all waves done reading buffer before overwrite
    if (lead && (c + NBUF < C)) {
      tdm_load_tile<N>((unsigned)(uintptr_t)&sbuf[c & (NBUF - 1)][0],
                       x + base + (size_t)((c + NBUF) * TCHUNK * N));
    }
  }
}

// Generic fallback for unexpected shapes: direct coalesced scan, one thread
// per (batch, neuron) column.
__global__ void AssociativeLIF_fallback(const float* __restrict__ x,
                                        const float* __restrict__ tau_mem,
                                        const float* __restrict__ v_threshold,
                                        float* __restrict__ out,
                                        int B, int T, int N) {
  long long idx = (long long)blockIdx.x * blockDim.x + threadIdx.x;
  if (idx >= (long long)B * N) return;
  int b = (int)(idx / N);
  int n = (int)(idx - (long long)b * N);
  float tau = fminf(fmaxf(tau_mem[n], 0.8f), 0.98f);
  float thr = fminf(fmaxf(v_threshold[n], 0.05f), 0.5f);
  float omt = 1.0f - tau;
  float isyn = 0.0f, v = 0.0f;
  size_t p = (size_t)b * T * N + n;
  for (int t = 0; t < T; ++t, p += N) {
    float xv = x[p];
    isyn = fmaf(0.5f, isyn, xv);
    v    = fmaf(tau, v, omt * isyn);
    __builtin_nontemporal_store((v >= thr) ? 1.0f : 0.0f, &out[p]);
  }
}

extern "C" void kernel_launch(void* const* d_in, const int* in_sizes, int n_in,
                              void* d_out, int out_size, void* d_ws, size_t ws_size,
                              hipStream_t stream) {
  const float* x   = (const float*)d_in[0];   // (B,T,N) f32
  const float* tau = (const float*)d_in[1];   // (N,)    f32
  const float* thr = (const float*)d_in[2];   // (N,)    f32
  float* out = (float*)d_out;                 // (B,T,N) f32

  const int N = in_sizes[1];                  // 4096
  const int T = 128;
  const int B = in_sizes[0] / (T * N);        // 32

  if (N == 4096) {
    dim3 grid(N / COLS, B);                   // 16 x 32 = 512 workgroups
    AssociativeLIF_kernel<128, 4096><<<grid, COLS, 0, stream>>>(x, tau, thr, out);
  } else {
    long long cols = (long long)B * N;
    int blocks = (int)((cols + 255) / 256);
    AssociativeLIF_fallback<<<blocks, 256, 0, stream>>>(x, tau, thr, out, B, T, N);
  }
}